// AGCBlock_19439021982280
// MI455X (gfx1250) — compile-verified
//
#include <hip/hip_runtime.h>

// ---- dims from the reference ----
#define IMG 512
#define C_CH 64
#define P_CH 32
#define HW   (IMG * IMG)        // 262144 floats per channel
#define HW4  (HW / 4)           // 65536 float4 per channel == 2^16
#define EPS  1e-5f

typedef __attribute__((ext_vector_type(2))) float v2f;
typedef __attribute__((ext_vector_type(8))) float v8f;

// ---------------------------------------------------------------------------
// Kernel 1: one wave32. Computes term[c] = relu(LN(b1)*gamma+beta) @ w2^T + b2
// using V_WMMA_F32_16X16X4_F32 (full fp32 precision on the tensor pipe).
//   A (16x4): row M=0 = t[k..k+3]; lane0 holds K=0,1 ; lane16 holds K=2,3.
//   B (4x16): lane%16 = N ; v0/v1 = K pair (lanes>=16 use K+2).
//   D (16x16): row M=0 lives in element 0 of lanes 0..15.
// ---------------------------------------------------------------------------
__global__ __launch_bounds__(32) void agc_term_kernel(
    const float* __restrict__ b1,    // [P]
    const float* __restrict__ gamma, // [P]
    const float* __restrict__ beta,  // [P]
    const float* __restrict__ w2,    // [C, P] row-major
    const float* __restrict__ b2,    // [C]
    float* __restrict__ term)        // [C] out (workspace)
{
    const int lane = threadIdx.x;            // 0..31, wave32
    float t = b1[lane];                      // context is exactly zero -> t = b1

    // mean over P=32 (one full wave)
    float s = t;
    for (int off = 16; off > 0; off >>= 1) s += __shfl_xor(s, off, 32);
    const float mu = s * (1.0f / 32.0f);

    // biased variance
    const float d = t - mu;
    float v = d * d;
    for (int off = 16; off > 0; off >>= 1) v += __shfl_xor(v, off, 32);
    const float var = v * (1.0f / 32.0f);

    // LayerNorm affine + ReLU
    const float tl = fmaxf(d / sqrtf(var + EPS) * gamma[lane] + beta[lane], 0.0f);

    __shared__ float ts[P_CH];
    ts[lane] = tl;
    __syncthreads();   // single wave: LDS ops are in-order, barrier is a NOP

    const int nIdx  = lane & 15;             // N within a 16-wide tile
    const int kHalf = (lane >> 4) ? 2 : 0;   // upper half-wave carries K+2

    // 4 column tiles of 16 cover C=64; 8 chunks of K=4 cover P=32.
    for (int ct = 0; ct < 4; ++ct) {
        v8f acc = {};
        const int c = ct * 16 + nIdx;
#pragma unroll
        for (int kk = 0; kk < 8; ++kk) {
            const int kb = kk * 4;
            v2f a;
            a.x = 0.0f; a.y = 0.0f;
            if (lane == 0)  { a.x = ts[kb + 0]; a.y = ts[kb + 1]; }
            if (lane == 16) { a.x = ts[kb + 2]; a.y = ts[kb + 3]; }
            v2f b;
            b.x = w2[c * P_CH + kb + kHalf + 0];
            b.y = w2[c * P_CH + kb + kHalf + 1];
            // 8-arg form: (neg_a, A, neg_b, B, c_mod, C, reuse_a, reuse_b)
            acc = __builtin_amdgcn_wmma_f32_16x16x4_f32(
                false, a, false, b, (short)0, acc, false, false);
        }
        if (lane < 16) {
            term[c] = acc[0] + b2[c];        // row M=0 of D
        }
    }
}

// ---------------------------------------------------------------------------
// Kernel 2: pure-bandwidth elementwise out = x + term[c].
// fold(unfold(x)+term)/counts == x + term[c] (all pixels covered; counts>=1).
// float4 path -> global_load_b128 / global_store_b128; ~128 MiB total traffic.
// ---------------------------------------------------------------------------
__global__ __launch_bounds__(256) void agc_add_kernel(
    const float4* __restrict__ x,
    const float* __restrict__ term,
    float4* __restrict__ out)
{
    const unsigned i4 = blockIdx.x * 256u + threadIdx.x;   // 0 .. 4194303
    const int c = (int)(i4 >> 16);                         // HW4 == 2^16
    const float tc = term[c];                              // block-uniform load
    float4 val = x[i4];
    val.x += tc; val.y += tc; val.z += tc; val.w += tc;
    out[i4] = val;
}

// ---------------------------------------------------------------------------
// inputs: 0:x 1:w_mask 2:b_mask 3:w1 4:b1 5:gamma 6:beta 7:w2 8:b2 (all f32)
// w_mask/b_mask/w1 are mathematically dead (log_softmax over a singleton axis
// is exactly zero, annihilating the context term).
// ---------------------------------------------------------------------------
extern "C" void kernel_launch(void* const* d_in, const int* in_sizes, int n_in,
                              void* d_out, int out_size, void* d_ws, size_t ws_size,
                              hipStream_t stream) {
    const float* x     = (const float*)d_in[0];
    const float* b1    = (const float*)d_in[4];
    const float* gamma = (const float*)d_in[5];
    const float* beta  = (const float*)d_in[6];
    const float* w2    = (const float*)d_in[7];
    const float* b2    = (const float*)d_in[8];

    float* term = (float*)d_ws;   // 64 floats of scratch

    agc_term_kernel<<<1, 32, 0, stream>>>(b1, gamma, beta, w2, b2, term);

    const int total4 = (C_CH * HW) / 4;        // 4,194,304 float4
    agc_add_kernel<<<total4 / 256, 256, 0, stream>>>(
        (const float4*)x, term, (float4*)d_out);
}